// FixedPooling2D_71201967833962
// MI455X (gfx1250) — compile-verified
//
#include <hip/hip_runtime.h>
#include <cstdint>
#include <cstddef>

// ---------------------------------------------------------------------------
// Device helpers
// ---------------------------------------------------------------------------
__device__ __forceinline__ float4 fmax4(float4 a, float4 b) {
    float4 r;
    r.x = fmaxf(a.x, b.x);
    r.y = fmaxf(a.y, b.y);
    r.z = fmaxf(a.z, b.z);
    r.w = fmaxf(a.w, b.w);
    return r;
}

typedef int v4i_ext __attribute__((ext_vector_type(4)));
typedef __attribute__((address_space(1))) v4i_ext* g_v4i_ptr;
typedef __attribute__((address_space(3))) v4i_ext* l_v4i_ptr;

// Async global -> LDS copy of 16 bytes per lane (CDNA5 ASYNCcnt path).
// Builtin signature (probe-confirmed by round-1 diagnostic):
//   void __builtin_amdgcn_global_load_async_to_lds_b128(v4i AS1* src, v4i AS3* dst,
//                                                       imm int offset, imm int cpol)
__device__ __forceinline__ void async_copy_b128(uint32_t lds_off, uint64_t gaddr) {
#if __has_builtin(__builtin_amdgcn_global_load_async_to_lds_b128)
    __builtin_amdgcn_global_load_async_to_lds_b128(
        (g_v4i_ptr)(uintptr_t)gaddr,
        (l_v4i_ptr)(uintptr_t)lds_off,
        0, 0);
#else
    asm volatile("global_load_async_to_lds_b128 %0, %1, off"
                 :: "v"(lds_off), "v"(gaddr)
                 : "memory");
#endif
}

template <int N>
__device__ __forceinline__ void wait_asynccnt() {
#if __has_builtin(__builtin_amdgcn_s_wait_asynccnt)
    __builtin_amdgcn_s_wait_asynccnt(N);
#else
    asm volatile("s_wait_asynccnt %0" :: "n"(N) : "memory");
#endif
}

// ---------------------------------------------------------------------------
// Kernel 1: 32x32 stride-32 SAME max pool, NHWC [32,225,225,128] -> [32,8,8,128]
// One 256-thread block per pooled cell (b,i,j). Rows of the window are streamed
// through LDS with double-buffered async copies.
// ---------------------------------------------------------------------------
#define H 225
#define W 225
#define C 128
#define ROWBUF_F4 1024   // 32 cols * 32 float4 (128 ch) = 16 KB per buffer

__global__ __launch_bounds__(256)
void pool_max_kernel(const float* __restrict__ x, float* __restrict__ pooled) {
    __shared__ float4 buf[2][ROWBUF_F4];   // 32 KB
    __shared__ float4 red[256];            // 4 KB

    const int tid  = threadIdx.x;
    const int cell = blockIdx.x;           // b*64 + i*8 + j
    const int b = cell >> 6;
    const int i = (cell >> 3) & 7;
    const int j = cell & 7;

    const int r0 = max(0, i * 32 - 15);
    const int r1 = min(H, i * 32 + 17);
    const int c0 = max(0, j * 32 - 15);
    const int c1 = min(W, j * 32 + 17);
    const int nrows = r1 - r0;             // 16..32
    const int ncols = c1 - c0;             // 16..32
    const int rowBytes = ncols * C * 4;    // <= 16384, multiple of 512

    const uint64_t rowStrideB = (uint64_t)W * C * 4;
    const uint64_t base = (uint64_t)(uintptr_t)x
                        + (uint64_t)b * H * rowStrideB
                        + (uint64_t)c0 * C * 4;

    const uint32_t lds0 = (uint32_t)(uintptr_t)(&buf[0][0]);
    const uint32_t lds1 = (uint32_t)(uintptr_t)(&buf[1][0]);

    auto issue_row = [&](int r, int which) {
        const uint64_t g = base + (uint64_t)r * rowStrideB;
        const uint32_t l = which ? lds1 : lds0;
#pragma unroll
        for (int k = 0; k < 4; ++k) {
            int off  = (tid + k * 256) * 16;                   // 0..16368
            int offc = off < rowBytes ? off : (rowBytes - 16); // clamp ragged tail
            async_copy_b128(l + (uint32_t)off, g + (uint64_t)offc);
        }
    };

    const int cg = tid & 31;   // channel quad (float4 of channels 4*cg..4*cg+3)
    const int s  = tid >> 5;   // wave id 0..7 -> cols s, s+8, s+16, s+24

    float4 acc;
    acc.x = acc.y = acc.z = acc.w = -__builtin_inff();

    issue_row(r0, 0);
    for (int rr = 0; rr < nrows; ++rr) {
        const int cur = rr & 1;
        if (rr + 1 < nrows) {
            issue_row(r0 + rr + 1, cur ^ 1);
            wait_asynccnt<4>();   // allow the 4 just-issued -> current row landed
        } else {
            wait_asynccnt<0>();
        }
        __syncthreads();          // all waves' copies of buf[cur] visible
#pragma unroll
        for (int k = 0; k < 4; ++k) {
            const int col = s + k * 8;
            if (col < ncols) acc = fmax4(acc, buf[cur][col * 32 + cg]);
        }
        __syncthreads();          // done reading buf[cur] before it is refilled
    }

    // cross-wave max over the 8 waves sharing this channel quad
    red[tid] = acc;
    __syncthreads();
    if (s == 0) {
        float4 m = red[cg];
#pragma unroll
        for (int w = 1; w < 8; ++w) m = fmax4(m, red[w * 32 + cg]);
        reinterpret_cast<float4*>(pooled)[(size_t)cell * 32 + cg] = m;
    }
}

// ---------------------------------------------------------------------------
// Kernel 2: ten static row/col pair-merges of pooled [32,8,8,128], averaged.
// out[b,oh,ow,c] = 0.1 * sum_it max over {rows(oh,i_it)} x {cols(ow,j_it)}
// ---------------------------------------------------------------------------
struct MergeIdx { int ri[10]; int ci[10]; };

__global__ __launch_bounds__(256)
void merge_avg_kernel(const float* __restrict__ pooled, float* __restrict__ out,
                      MergeIdx idx) {
    const int t = blockIdx.x * 256 + threadIdx.x;   // 32*7*7*32 = 50176 float4s
    if (t >= 32 * 7 * 7 * 32) return;
    const int cg = t & 31;
    const int q  = t >> 5;          // (b*7+oh)*7+ow
    const int ow = q % 7;
    const int oh = (q / 7) % 7;
    const int b  = q / 49;

    const float4* p = reinterpret_cast<const float4*>(pooled);
    float4 acc; acc.x = acc.y = acc.z = acc.w = 0.0f;

#pragma unroll
    for (int it = 0; it < 10; ++it) {
        const int i = idx.ri[it];
        const int j = idx.ci[it];
        const int r = oh + (oh > i ? 1 : 0);
        const int c = ow + (ow > j ? 1 : 0);
        const size_t base = ((size_t)(b * 8 + r) * 8 + c) * 32 + cg;
        float4 m = p[base];
        if (oh == i)            m = fmax4(m, p[base + 8 * 32]);
        if (ow == j)            m = fmax4(m, p[base + 32]);
        if (oh == i && ow == j) m = fmax4(m, p[base + 8 * 32 + 32]);
        acc.x += m.x; acc.y += m.y; acc.z += m.z; acc.w += m.w;
    }
    float4 o;
    o.x = acc.x * 0.1f; o.y = acc.y * 0.1f; o.z = acc.z * 0.1f; o.w = acc.w * 0.1f;
    reinterpret_cast<float4*>(out)[t] = o;
}

// ---------------------------------------------------------------------------
// Host: replicate numpy RandomState(42+it).randint(0,7) draws (MT19937, legacy
// masked-rejection bounded int64: next_uint64 = (next32<<32)|next32, mask 7,
// reject 7).
// ---------------------------------------------------------------------------
namespace {

struct MT19937Host {
    uint32_t mt[624];
    int mti;
    void seed(uint32_t s) {
        mt[0] = s;
        for (mti = 1; mti < 624; ++mti)
            mt[mti] = 1812433253u * (mt[mti - 1] ^ (mt[mti - 1] >> 30)) + (uint32_t)mti;
    }
    uint32_t next() {
        if (mti >= 624) {
            for (int k = 0; k < 624; ++k) {
                uint32_t y = (mt[k] & 0x80000000u) | (mt[(k + 1) % 624] & 0x7fffffffu);
                uint32_t v = mt[(k + 397) % 624] ^ (y >> 1);
                if (y & 1u) v ^= 2567483615u;
                mt[k] = v;
            }
            mti = 0;
        }
        uint32_t y = mt[mti++];
        y ^= y >> 11;
        y ^= (y << 7) & 2636928640u;
        y ^= (y << 15) & 4022730752u;
        y ^= y >> 18;
        return y;
    }
    uint64_t next64() {
        uint64_t hi = next();
        uint64_t lo = next();
        return (hi << 32) | lo;
    }
    int randint7() {  // numpy legacy randint(0, 7)
        for (;;) {
            uint64_t v = next64() & 7ull;
            if (v <= 6ull) return (int)v;
        }
    }
};

}  // namespace

extern "C" void kernel_launch(void* const* d_in, const int* in_sizes, int n_in,
                              void* d_out, int out_size, void* d_ws, size_t ws_size,
                              hipStream_t stream) {
    (void)in_sizes; (void)n_in; (void)out_size; (void)ws_size;

    const float* x   = (const float*)d_in[0];
    float*       out = (float*)d_out;
    float*       pooled = (float*)d_ws;    // 32*8*8*128*4 = 1 MB scratch

    MergeIdx idx;
    for (int it = 0; it < 10; ++it) {
        MT19937Host g;
        g.seed(42u + (uint32_t)it);
        idx.ri[it] = g.randint7();   // row merge index (first draw)
        idx.ci[it] = g.randint7();   // col merge index (second draw)
    }

    pool_max_kernel<<<32 * 8 * 8, 256, 0, stream>>>(x, pooled);

    const int total = 32 * 7 * 7 * 32;   // float4 outputs
    merge_avg_kernel<<<(total + 255) / 256, 256, 0, stream>>>(pooled, out, idx);
}